// ELM_AE_Spatial_Pool2d_36490042147430
// MI455X (gfx1250) — compile-verified
//
#include <hip/hip_runtime.h>
#include <cstdint>

#define BATCH   256
#define ZCH     2048
#define ROWLEN  49            // 7*7 floats per (b,z) map
#define TILE_A  32            // z-rows per stats tile
#define NSTEP_A (ZCH / TILE_A)
#define TILE_B  256           // z-rows per main-kernel block

// ---------------- feature detection (host pass falls back harmlessly) ----
#if __has_builtin(__builtin_amdgcn_tensor_load_to_lds) && __has_builtin(__builtin_amdgcn_s_wait_tensorcnt)
#define HAVE_TDM 1
#else
#define HAVE_TDM 0
#endif
#if __has_builtin(__builtin_amdgcn_wmma_f32_16x16x4_f32)
#define HAVE_WMMA 1
#else
#define HAVE_WMMA 0
#endif

typedef unsigned int u32x4 __attribute__((ext_vector_type(4)));
typedef int          i32x8 __attribute__((ext_vector_type(8)));
typedef int          i32x4 __attribute__((ext_vector_type(4)));
typedef float        v2f   __attribute__((ext_vector_type(2)));
typedef float        v8f   __attribute__((ext_vector_type(8)));

// 7 -> 8 bilinear tap: matches jax.image.resize linear weights with edge
// normalization (i=0 -> pixel 0, i=7 -> pixel 6, interior = lerp(floor, frac)).
__device__ __forceinline__ void rw7to8(int i, int& i0, float& w) {
    float f = 0.875f * (float)i - 0.0625f;
    f = fminf(fmaxf(f, 0.0f), 6.0f);
    int t = (int)f;
    t = (t > 5) ? 5 : t;
    i0 = t;
    w  = f - (float)t;
}

__device__ __forceinline__ float bilin(const float* r, int y0, int x0, float wx, float wy) {
    const float* p = r + (y0 * 7 + x0);          // y0 <= 5, x0 <= 5 -> all reads in-bounds
    float a0 = p[0], a1 = p[1];
    float b0 = p[7], b1 = p[8];
    float h0 = a0 + wx * (a1 - a0);
    float h1 = b0 + wx * (b1 - b0);
    return h0 + wy * (h1 - h0);
}

__device__ __forceinline__ float bilr(const float* raw, int oy, int ox) {
    int y0, x0; float wy, wx;
    rw7to8(oy, y0, wy);
    rw7to8(ox, x0, wx);
    return bilin(raw, y0, x0, wx, wy);
}

#if HAVE_TDM
// 1-D contiguous TDM tile load: nelem f32 from global -> LDS.
// D# per CDNA5 ISA: group0 {count=1, lds_addr, global_addr, type=2},
// group1 {data_size=4B, tensor_dim0=tile_dim0=stride0=nelem, tensor_dim1=tile_dim1=1}.
// clang-23 toolchain builtin arity is 6: (u32x4, i32x8, i32x4, i32x4, i32x8, i32 cpol);
// groups 2/3 (and the trailing group) are zero for a <=2-D tensor.
__device__ __forceinline__ void tdm_load_1d(unsigned lds_addr, const void* gsrc, unsigned nelem) {
    unsigned long long ga = (unsigned long long)(uintptr_t)gsrc;
    u32x4 g0;
    g0.x = 1u;                                                  // count = 1 (valid)
    g0.y = lds_addr;                                            // LDS byte address
    g0.z = (unsigned)ga;                                        // global_addr[31:0]
    g0.w = ((unsigned)(ga >> 32) & 0x01FFFFFFu) | 0x80000000u;  // addr[56:32] | type=2
    const unsigned rl = nelem, nr = 1u, span = nelem;
    i32x8 g1;
    g1[0] = 0x00020000;                                         // data_size = 4 bytes
    g1[1] = (int)((rl & 0xFFFFu) << 16);                        // tensor_dim0 lo16
    g1[2] = (int)(((rl >> 16) & 0xFFFFu) | ((nr & 0xFFFFu) << 16)); // dim0 hi | dim1 lo
    g1[3] = (int)(((nr >> 16) & 0xFFFFu) | ((rl & 0xFFFFu) << 16)); // dim1 hi | tile_dim0
    g1[4] = (int)(nr & 0xFFFFu);                                // tile_dim1 | tile_dim2=0
    g1[5] = (int)rl;                                            // tensor_dim0_stride lo32
    g1[6] = (int)((span & 0xFFFFu) << 16);                      // stride0 hi | stride1 lo
    g1[7] = (int)(span >> 16);                                  // stride1 hi
    i32x4 z4 = {0, 0, 0, 0};
    i32x8 z8 = {0, 0, 0, 0, 0, 0, 0, 0};
    __builtin_amdgcn_tensor_load_to_lds(g0, g1, z4, z4, z8, 0);
}
#endif

// ------------------------------------------------------------------------
// Pass 1: per (b, pixel) mean / 1/std over z (2048 channels).
// WMMA ones-matmul reduction: A[m,k] = resized value (pixel m, z-row k),
// B = ones  =>  D[m,n] = sum_k A[m,k] for every n.  C accumulates.
// ------------------------------------------------------------------------
__global__ void __launch_bounds__(256) radam_stats_kernel(const float* __restrict__ x,
                                                          float* __restrict__ meanArr,
                                                          float* __restrict__ rstdArr) {
    __shared__ float tile[2][TILE_A * ROWLEN];
    __shared__ float part[16][2][64];

    const int b      = blockIdx.x;
    const int tid    = threadIdx.x;
    const int lane   = tid & 31;
    const int wv     = tid >> 5;          // 8 waves, wave wv owns rows 4*wv..4*wv+3 of each tile
    const int m      = lane & 15;         // WMMA A-matrix M index (pixel within 16-group)
    const int lhalf  = lane >> 4;
    const int k0     = lhalf * 2;         // lane holds K = k0, k0+1
    const int ox     = m & 7;
    const int oyb    = m >> 3;
    int x0; float wx; rw7to8(ox, x0, wx);

    const float* src = x + (size_t)b * (ZCH * ROWLEN);

#if HAVE_WMMA
    v8f csum[4], csq[4];
    const v2f onesv = {1.0f, 1.0f};
    #pragma unroll
    for (int g = 0; g < 4; ++g) {
        csum[g] = (v8f){0,0,0,0,0,0,0,0};
        csq[g]  = (v8f){0,0,0,0,0,0,0,0};
    }
#else
    float ps[4] = {0,0,0,0}, pq[4] = {0,0,0,0};
#endif

    // prologue: tile 0
#if HAVE_TDM
    if (tid == 0) tdm_load_1d((unsigned)(uintptr_t)&tile[0][0], src, TILE_A * ROWLEN);
    if (tid < 32) __builtin_amdgcn_s_wait_tensorcnt(0);
    __syncthreads();
#else
    for (int i = tid; i < TILE_A * ROWLEN; i += 256) tile[0][i] = src[i];
    __syncthreads();
#endif

    for (int step = 0; step < NSTEP_A; ++step) {
        const int cur = step & 1;
#if HAVE_TDM
        if (tid == 0 && step + 1 < NSTEP_A)   // prefetch next tile while computing
            tdm_load_1d((unsigned)(uintptr_t)&tile[1 - cur][0],
                        src + (size_t)(step + 1) * TILE_A * ROWLEN, TILE_A * ROWLEN);
#endif
        const float* r0 = &tile[cur][(wv * 4 + k0) * ROWLEN];
        float A0[4], A1[4];
        #pragma unroll
        for (int g = 0; g < 4; ++g) {
            int oy = 2 * g + oyb;             // pixel index = g*16 + m  <->  (oy, ox)
            int y0; float wy; rw7to8(oy, y0, wy);
            A0[g] = bilin(r0,          y0, x0, wx, wy);   // K = k0
            A1[g] = bilin(r0 + ROWLEN, y0, x0, wx, wy);   // K = k0 + 1
        }
#if HAVE_WMMA
        #pragma unroll
        for (int g = 0; g < 4; ++g) {
            v2f a  = {A0[g], A1[g]};
            v2f aq = {A0[g] * A0[g], A1[g] * A1[g]};
            csum[g] = __builtin_amdgcn_wmma_f32_16x16x4_f32(false, a,  false, onesv,
                                                            (short)0, csum[g], false, false);
            csq[g]  = __builtin_amdgcn_wmma_f32_16x16x4_f32(false, aq, false, onesv,
                                                            (short)0, csq[g],  false, false);
        }
#else
        #pragma unroll
        for (int g = 0; g < 4; ++g) {
            ps[g] += A0[g] + A1[g];
            pq[g] += A0[g] * A0[g] + A1[g] * A1[g];
        }
#endif
#if HAVE_TDM
        if (tid < 32) __builtin_amdgcn_s_wait_tensorcnt(0);
        __syncthreads();
#else
        __syncthreads();
        if (step + 1 < NSTEP_A) {
            const float* nsrc = src + (size_t)(step + 1) * TILE_A * ROWLEN;
            for (int i = tid; i < TILE_A * ROWLEN; i += 256) tile[1 - cur][i] = nsrc[i];
            __syncthreads();
        }
#endif
    }

    // stash per-wave partials: D layout (f32 16x16): VGPR v, lanes<16 -> M=v, lanes>=16 -> M=v+8
#if HAVE_WMMA
    if (lane == 0 || lane == 16) {
        const int mb = lhalf * 8;
        #pragma unroll
        for (int g = 0; g < 4; ++g) {
            #pragma unroll
            for (int v = 0; v < 8; ++v) {
                const int px = g * 16 + mb + v;
                part[wv][0][px] = csum[g][v];
                part[wv][1][px] = csq[g][v];
            }
        }
    }
    const int NS = 8;
#else
    #pragma unroll
    for (int g = 0; g < 4; ++g) {
        part[wv * 2 + lhalf][0][g * 16 + m] = ps[g];
        part[wv * 2 + lhalf][1][g * 16 + m] = pq[g];
    }
    const int NS = 16;
#endif
    __syncthreads();
    if (tid < 64) {
        float S = 0.f, Q = 0.f;
        for (int s = 0; s < NS; ++s) { S += part[s][0][tid]; Q += part[s][1][tid]; }
        const float mean = S * (1.0f / ZCH);
        const float var  = fmaxf(Q * (1.0f / ZCH) - mean * mean, 0.0f);
        const float sd   = sqrtf(var);
        meanArr[b * 64 + tid] = mean;
        rstdArr[b * 64 + tid] = (sd > 0.0f) ? (1.0f / sd) : 0.0f;   // nan_to_num(0/0)=0
    }
}

// ------------------------------------------------------------------------
// Pass 2: one z-row per thread -> zscore(z), zscore(patch), ELM, beta, std.
// ------------------------------------------------------------------------
__global__ void __launch_bounds__(256) radam_main_kernel(const float* __restrict__ x,
                                                         const float* __restrict__ W,
                                                         const float* __restrict__ meanArr,
                                                         const float* __restrict__ rstdArr,
                                                         float* __restrict__ out) {
    __shared__ float tile[TILE_B * ROWLEN];
    __shared__ float sm[64];
    __shared__ float sr[64];

    const int bid = blockIdx.x;
    const int b   = bid >> 3;
    const int tz  = bid & 7;
    const int tid = threadIdx.x;
    const float* src = x + ((size_t)b * ZCH + (size_t)tz * TILE_B) * ROWLEN;

#if HAVE_TDM
    if (tid == 0) tdm_load_1d((unsigned)(uintptr_t)&tile[0], src, TILE_B * ROWLEN);
#else
    for (int i = tid; i < TILE_B * ROWLEN; i += 256) tile[i] = src[i];
#endif
    if (tid < 64)        sm[tid]      = meanArr[b * 64 + tid];
    else if (tid < 128)  sr[tid - 64] = rstdArr[b * 64 + tid - 64];
#if HAVE_TDM
    if (tid < 32) __builtin_amdgcn_s_wait_tensorcnt(0);
#endif
    __syncthreads();

    // pull this thread's row into registers (stride 49 -> bank-conflict-free)
    float raw[ROWLEN];
    const float* rp = &tile[tid * ROWLEN];
    #pragma unroll
    for (int i = 0; i < ROWLEN; ++i) raw[i] = rp[i];

    const float w0 = W[0], w1 = W[1], w2 = W[2], w3 = W[3], w4 = W[4];

    float zz = 0.f;
    float bacc[4] = {0, 0, 0, 0};
    #pragma unroll
    for (int gy = 0; gy < 4; ++gy) {
        #pragma unroll
        for (int gx = 0; gx < 4; ++gx) {
            float Y[4];
            #pragma unroll
            for (int dy = 0; dy < 2; ++dy) {
                #pragma unroll
                for (int dx = 0; dx < 2; ++dx) {
                    const int oy = 2 * gy + dy, oxx = 2 * gx + dx;
                    const int p8 = oy * 8 + oxx;
                    const float R = bilr(raw, oy, oxx);       // indices fold at compile time
                    Y[dy * 2 + dx] = (R - sm[p8]) * sr[p8];   // zscore over z (nan_to_num)
                }
            }
            // zscore over the 4 patch pixels (ddof=0)
            const float mu = 0.25f * (Y[0] + Y[1] + Y[2] + Y[3]);
            const float d0 = Y[0] - mu, d1 = Y[1] - mu, d2 = Y[2] - mu, d3 = Y[3] - mu;
            const float var = 0.25f * (d0 * d0 + d1 * d1 + d2 * d2 + d3 * d3);
            const float sd  = sqrtf(var);
            const float inv = (sd > 0.f) ? (1.0f / sd) : 0.0f;
            const float X0 = d0 * inv, X1 = d1 * inv, X2 = d2 * inv, X3 = d3 * inv;
            // ELM hidden: sigmoid(W . [1; Xc])
            const float hpre = w0 + w1 * X0 + w2 * X1 + w3 * X2 + w4 * X3;
            const float hid  = 1.0f / (1.0f + __expf(-hpre));
            zz += hid * hid;
            bacc[0] += X0 * hid; bacc[1] += X1 * hid; bacc[2] += X2 * hid; bacc[3] += X3 * hid;
        }
    }
    const float izz = 1.0f / zz;                 // zz > 0 (sigmoid > 0)
    const float b0 = bacc[0] * izz, b1 = bacc[1] * izz, b2 = bacc[2] * izz, b3 = bacc[3] * izz;
    const float mb = 0.25f * (b0 + b1 + b2 + b3);
    const float e0 = b0 - mb, e1 = b1 - mb, e2 = b2 - mb, e3 = b3 - mb;
    out[(size_t)b * ZCH + tz * TILE_B + tid] =
        sqrtf(0.25f * (e0 * e0 + e1 * e1 + e2 * e2 + e3 * e3));
}

extern "C" void kernel_launch(void* const* d_in, const int* in_sizes, int n_in,
                              void* d_out, int out_size, void* d_ws, size_t ws_size,
                              hipStream_t stream) {
    const float* x = (const float*)d_in[0];
    const float* W = (const float*)d_in[1];
    float* out     = (float*)d_out;
    float* meanArr = (float*)d_ws;               // BATCH*64 floats
    float* rstdArr = meanArr + BATCH * 64;       // BATCH*64 floats (128 KB of ws total)

    radam_stats_kernel<<<BATCH, 256, 0, stream>>>(x, meanArr, rstdArr);
    radam_main_kernel<<<BATCH * (ZCH / TILE_B), 256, 0, stream>>>(x, W, meanArr, rstdArr, out);

    (void)in_sizes; (void)n_in; (void)out_size; (void)ws_size;
}